// myloss_16862041604208
// MI455X (gfx1250) — compile-verified
//
#include <hip/hip_runtime.h>
#include <math.h>

typedef float v2f __attribute__((ext_vector_type(2)));
typedef float v8f __attribute__((ext_vector_type(8)));

#define D      512
#define KCLU   16
#define EPSPD  1e-6f
#define EPSCOS 1e-8f
#define EMAX   640   // padded e-extent (E=625 at runtime)

// ws layout (float offsets)
#define WS_ACC 0                         // 4 floats (only [0] used)
#define WS_TN  16                        // 16*16
#define WS_FN  (16 + KCLU*KCLU)          // 16*EMAX
#define WS_RAW (WS_FN + KCLU*EMAX)       // 16*16*EMAX

__global__ void init_kernel(float* ws) {
    if (threadIdx.x == 0) ws[WS_ACC] = 0.0f;
}

// ---------------- inner cluster loss: one wave32 per node ----------------
// Bandwidth-critical kernel (~205 MB stream): float4 per lane -> b128 loads.
__global__ void inner_kernel(const float* __restrict__ node,
                             const int* __restrict__ lab,
                             const float* __restrict__ cen,
                             const unsigned char* __restrict__ mask,
                             const float* __restrict__ mw,
                             float* __restrict__ acc, int N) {
    const int wave = threadIdx.x >> 5;
    const int lane = threadIdx.x & 31;
    const int nid  = blockIdx.x * 8 + wave;
    float contrib = 0.0f;
    if (nid < N) {
        const float4* x4 = (const float4*)(node + (size_t)nid * D);
        const float4* c4 = (const float4*)(cen + (size_t)lab[nid] * D);
        float s = 0.0f;
        for (int t = lane; t < D / 4; t += 32) {   // 4 iterations, b128 each
            const float4 xv = x4[t];
            const float4 cv = c4[t];
            float v;
            v = xv.x - cv.x + EPSPD; s += v * v;
            v = xv.y - cv.y + EPSPD; s += v * v;
            v = xv.z - cv.z + EPSPD; s += v * v;
            v = xv.w - cv.w + EPSPD; s += v * v;
        }
        for (int off = 16; off; off >>= 1) s += __shfl_xor(s, off, 32);
        if (lane == 0) {
            float dn = sqrtf(s);
            contrib = dn + (mask[nid] ? (1.0f + mw[0]) * dn : 0.0f);
        }
    }
    __shared__ float bs[8];
    if (lane == 0) bs[wave] = contrib;
    __syncthreads();
    if (threadIdx.x == 0) {
        float t = 0.0f;
        for (int w = 0; w < 8; ++w) t += bs[w];
        atomicAdd(acc, t);
    }
}

// ------------- norms: Fn[i,e] (edge row norms) and Tn[i,j] ---------------
__global__ void norms_kernel(const float* __restrict__ node,
                             const float* __restrict__ cen,
                             const int* __restrict__ sidx,
                             float* __restrict__ ws, int M, int E) {
    const int b    = blockIdx.x;
    const int lane = threadIdx.x;
    const int nF   = KCLU * E;
    float s = 0.0f;
    if (b < nF) {
        const int i = b / E, e = b % E;
        const int row = sidx[(size_t)i * M + (M - E) + e];
        const float4* x4 = (const float4*)(node + (size_t)row * D);
        for (int t = lane; t < D / 4; t += 32) {
            const float4 xv = x4[t];
            s += xv.x * xv.x + xv.y * xv.y + xv.z * xv.z + xv.w * xv.w;
        }
        for (int off = 16; off; off >>= 1) s += __shfl_xor(s, off, 32);
        if (lane == 0) ws[WS_FN + i * EMAX + e] = fmaxf(sqrtf(s), EPSCOS);
    } else {
        const int p = b - nF;
        const int i = p >> 4, j = p & 15;
        const float4* ci = (const float4*)(cen + (size_t)i * D);
        const float4* cj = (const float4*)(cen + (size_t)j * D);
        for (int t = lane; t < D / 4; t += 32) {
            const float4 a = ci[t];
            const float4 b2 = cj[t];
            float v;
            v = a.x - b2.x; s += v * v;
            v = a.y - b2.y; s += v * v;
            v = a.z - b2.z; s += v * v;
            v = a.w - b2.w; s += v * v;
        }
        for (int off = 16; off; off >>= 1) s += __shfl_xor(s, off, 32);
        if (lane == 0) ws[WS_TN + i * 16 + j] = fmaxf(sqrtf(s), EPSCOS);
    }
}

// ------ raw[i,j,e] = sum_d (c[i,d]-c[j,d]) * edge[i,e,d]  via WMMA fp32 ------
// One wave per (i, 16-wide e-tile). A = T[i] (16x512, M=j), B = gathered
// edge rows (512 x 16, N=e), K stepped by 4 with V_WMMA_F32_16X16X4_F32
// (exact fp32: the output feeds rank-q thresholding, so no downcast).
// Padded columns (e >= E) duplicate row E-1 and produce garbage raw values
// that are never read by pair_kernel -> no masking VALU work needed.
__global__ void wmma_cos_kernel(const float* __restrict__ node,
                                const float* __restrict__ cen,
                                const int* __restrict__ sidx,
                                float* __restrict__ ws, int M, int E) {
    const int i    = blockIdx.y;
    const int e0   = blockIdx.x * 16;
    const int lane = threadIdx.x;      // 0..31
    const int half = lane >> 4;        // 0 or 1
    const int l16  = lane & 15;
    const int kk   = half * 2;         // K sub-offset per ISA A/B fragment layout

    const float* ci = cen + (size_t)i   * D;   // A rows: T[i, m, :] = ci - cm
    const float* cm = cen + (size_t)l16 * D;

    const int e   = e0 + l16;
    const int ec  = (e < E) ? e : (E - 1);     // clamp; padded cols unread
    const int row = sidx[(size_t)i * M + (M - E) + ec];
    const float* bx = node + (size_t)row * D;

    v8f acc = {};
    for (int d = 0; d < D; d += 4) {
        v2f a, b;
        a.x = ci[d + kk]     - cm[d + kk];
        a.y = ci[d + kk + 1] - cm[d + kk + 1];
        b.x = bx[d + kk];
        b.y = bx[d + kk + 1];
        acc = __builtin_amdgcn_wmma_f32_16x16x4_f32(
            /*neg_a=*/false, a, /*neg_b=*/false, b,
            /*c_mod=*/(short)0, acc, /*reuse_a=*/false, /*reuse_b=*/false);
    }
    // D layout: VGPR r -> M = r + 8*half, N = l16
    float* raw = ws + WS_RAW;
#pragma unroll
    for (int r = 0; r < 8; ++r) {
        const int j = r + half * 8;
        raw[((size_t)(i * 16 + j)) * EMAX + e0 + l16] = acc[r];
    }
}

// ---- per-(i<j) pair: rank-q thresholds, keep counts, mean diff distance ----
__global__ void pair_kernel(const float* __restrict__ node,
                            const int* __restrict__ sidx,
                            float* __restrict__ ws, int M, int E, int q) {
    __shared__ float cosa[EMAX], cosb[EMAX];
    __shared__ int   rli[EMAX],  rlj[EMAX];
    __shared__ float tha_sh, thb_sh;
    __shared__ float red[256];
    __shared__ int   counts[2];

    // map blockIdx -> (i, j) with i < j
    int p = blockIdx.x;
    int i = 0;
    while (p >= KCLU - 1 - i) { p -= KCLU - 1 - i; ++i; }
    const int j = i + 1 + p;

    const int tid = threadIdx.x;
    const float* raw = ws + WS_RAW;
    const float  tn  = ws[WS_TN + i * 16 + j];

    for (int e = tid; e < E; e += blockDim.x) {
        rli[e] = sidx[(size_t)i * M + (M - E) + e];
        rlj[e] = sidx[(size_t)j * M + (M - E) + e];
        const float fni = ws[WS_FN + i * EMAX + e];
        const float fnj = ws[WS_FN + j * EMAX + e];
        cosa[e] =  raw[((size_t)(i * 16 + j)) * EMAX + e] / (tn * fni);
        // T[i,j] = -T[j,i]  =>  cos_b[i,j,e] = -raw[j,i,e]/(Tn*Fn[j,e])
        cosb[e] = -raw[((size_t)(j * 16 + i)) * EMAX + e] / (tn * fnj);
    }
    __syncthreads();

    // rank selection: th = sorted[q]  (element x with #less <= q < #leq)
    for (int e = tid; e < E; e += blockDim.x) {
        float x = cosa[e];
        int less = 0, leq = 0;
        for (int k = 0; k < E; ++k) { less += (cosa[k] < x); leq += (cosa[k] <= x); }
        if (less <= q && q < leq) tha_sh = x;   // racing writes carry identical value
        x = cosb[e];
        less = 0; leq = 0;
        for (int k = 0; k < E; ++k) { less += (cosb[k] < x); leq += (cosb[k] <= x); }
        if (less <= q && q < leq) thb_sh = x;
    }
    if (tid == 0) { counts[0] = 0; counts[1] = 0; }
    __syncthreads();
    const float tha = tha_sh, thb = thb_sh;

    int ca = 0, cb = 0;
    for (int e = tid; e < E; e += blockDim.x) {
        ca += (cosa[e] > tha);
        cb += (cosb[e] > thb);
    }
    atomicAdd(&counts[0], ca);
    atomicAdd(&counts[1], cb);
    __syncthreads();
    const float inv_a = 1.0f / fmaxf((float)counts[0], 1.0f);
    const float inv_b = 1.0f / fmaxf((float)counts[1], 1.0f);

    // || mean_a - mean_b + eps ||: thread owns d0 = 2*tid, float2 gathers
    const int d0 = tid * 2;                    // blockDim 256, D 512
    float sax = 0.0f, say = 0.0f, sbx = 0.0f, sby = 0.0f;
    for (int e = 0; e < E; ++e) {
        if (cosa[e] > tha) {
            const float2 v = *(const float2*)(node + (size_t)rli[e] * D + d0);
            sax += v.x; say += v.y;
        }
        if (cosb[e] > thb) {
            const float2 v = *(const float2*)(node + (size_t)rlj[e] * D + d0);
            sbx += v.x; sby += v.y;
        }
    }
    const float dx = sax * inv_a - sbx * inv_b + EPSPD;
    const float dy = say * inv_a - sby * inv_b + EPSPD;
    red[tid] = dx * dx + dy * dy;
    __syncthreads();
    for (int s = blockDim.x / 2; s > 0; s >>= 1) {
        if (tid < s) red[tid] += red[tid + s];
        __syncthreads();
    }
    if (tid == 0 && counts[0] > 0 && counts[1] > 0) {
        atomicAdd(ws + WS_ACC, -sqrtf(red[0]));
    }
}

__global__ void finalize_kernel(const float* __restrict__ ws, float* __restrict__ out) {
    out[0] = ws[WS_ACC];
}

extern "C" void kernel_launch(void* const* d_in, const int* in_sizes, int n_in,
                              void* d_out, int out_size, void* d_ws, size_t ws_size,
                              hipStream_t stream) {
    const float*         node = (const float*)d_in[0];
    const int*           lab  = (const int*)d_in[1];
    const float*         cen  = (const float*)d_in[2];
    const unsigned char* mask = (const unsigned char*)d_in[3]; // bool array (1B)
    const float*         mw   = (const float*)d_in[4];
    const int*           sidx = (const int*)d_in[5];
    float* ws  = (float*)d_ws;
    float* out = (float*)d_out;

    const int N = in_sizes[0] / D;        // 100000
    const int M = in_sizes[5] / KCLU;     // 6250
    const int E = M / 10;                 // 625  (EDGE_FRAC=0.1)
    const int q = E / 10;                 // 62

    init_kernel<<<1, 32, 0, stream>>>(ws);
    inner_kernel<<<(N + 7) / 8, 256, 0, stream>>>(node, lab, cen, mask, mw,
                                                  ws + WS_ACC, N);
    norms_kernel<<<KCLU * E + KCLU * KCLU, 32, 0, stream>>>(node, cen, sidx, ws, M, E);

    const int etiles = (E + 15) / 16;     // 40
    dim3 g(etiles, KCLU);
    wmma_cos_kernel<<<g, 32, 0, stream>>>(node, cen, sidx, ws, M, E);

    pair_kernel<<<KCLU * (KCLU - 1) / 2, 256, 0, stream>>>(node, sidx, ws, M, E, q);
    finalize_kernel<<<1, 1, 0, stream>>>(ws, out);
}